// Attention_78700980732135
// MI455X (gfx1250) — compile-verified
//
#include <hip/hip_runtime.h>
#include <hip/hip_bf16.h>
#include <math.h>

// Attention: scores = (cs . enc)/sqrt(D) + log(mask); dist = softmax_s; ctx = dist^T enc
// SEQ=2048, BATCH=32, DIM=1024, f32. HBM-bound (~0.5 FLOP/byte): enc read twice,
// all global traffic coalesced b128. Scores use V_WMMA_F32_16X16X4_F32 fed from LDS;
// enc tiles are staged with GLOBAL_LOAD_ASYNC_TO_LDS_B128 (ASYNCcnt) double-buffered
// against the WMMA loop.

#define SEQ   2048
#define BATCH 32
#define DIM   1024

typedef __attribute__((ext_vector_type(2))) float v2f;
typedef __attribute__((ext_vector_type(8))) float v8f;

// ---------------------------------------------------------------------------
// Kernel 1: weights[s,b] = dot(cs[b,:], enc[s,b,:]) / 32 + log(mask[s,b])
// One wave per (16-seq tile, batch). WMMA trick: A rows broadcast the key chunk,
// B columns hold 16 enc rows -> every C row carries the 16 scores.
// enc tile [16 x 128] double-buffered in LDS via async DMA (row stride 132 floats
// keeps the DS reads bank-conflict-free).
// ---------------------------------------------------------------------------
__global__ __launch_bounds__(32) void scores_kernel(
    const float* __restrict__ enc, const float* __restrict__ cs,
    const float* __restrict__ mask, float* __restrict__ wgt) {
  const int s0   = blockIdx.x * 16;
  const int b    = blockIdx.y;
  const int lane = threadIdx.x;        // 0..31, wave32
  const int n    = lane & 15;          // B/C column (= seq row within tile)
  const int h    = lane >> 4;          // K-half selector for A/B fragments

  __shared__ __align__(16) float keyL[DIM];            // whole key row: 4 KB
  __shared__ __align__(16) float encL[2][16 * 132];    // double-buffered tiles

  // stage key row once (coalesced b128 through VGPRs; tiny)
  {
    const float4* k4 = (const float4*)(cs + (size_t)b * DIM);
    for (int i = lane; i < DIM / 4; i += 32)
      *(float4*)&keyL[4 * i] = k4[i];
  }
  __syncthreads();

  // byte offset of the tile's row 0 within enc (fits in i32: enc is 256 MB)
  const unsigned gbase = ((unsigned)s0 * (BATCH * DIM) + (unsigned)b * DIM) * 4u;
  // generic->LDS: low 32 bits of a flat shared address are the wave LDS offset
  const unsigned lds0 = (unsigned)(size_t)&encL[0][0];
  const unsigned lds1 = (unsigned)(size_t)&encL[1][0];

  // issue one chunk: 16 async b128 loads per lane (lane covers float4 i, i+32, ...)
  auto issue = [&](unsigned ldsbase, int kc) {
    #pragma unroll
    for (int i = 0; i < 16; ++i) {
      const int idx = lane + 32 * i;     // 0..511 float4 of the 16x128 tile
      const int row = idx >> 5, c4 = idx & 31;
      unsigned ldsa = ldsbase + (unsigned)(row * 132 + 4 * c4) * 4u;
      unsigned voff = gbase + ((unsigned)row * (BATCH * DIM) + (unsigned)kc
                               + 4u * (unsigned)c4) * 4u;
      asm volatile("global_load_async_to_lds_b128 %0, %1, %2"
                   :: "v"(ldsa), "v"(voff), "s"(enc) : "memory");
    }
  };

  issue(lds0, 0);                        // prefetch chunk 0 -> buf 0 (ASYNCcnt=16)

  v8f c = {0.f, 0.f, 0.f, 0.f, 0.f, 0.f, 0.f, 0.f};

  int buf = 0;
  for (int kc = 0; kc < DIM; kc += 128, buf ^= 1) {
    if (kc + 128 < DIM) {
      issue(buf ? lds0 : lds1, kc + 128);              // overlap next chunk's DMA
      asm volatile("s_wait_asynccnt 0x10" ::: "memory"); // chunk kc resident
    } else {
      asm volatile("s_wait_asynccnt 0x0" ::: "memory");
    }

    const float* rowp = &encL[buf][n * 132];  // enc row for this lane's B column
    const float* keyp = &keyL[kc];
    #pragma unroll
    for (int k = 0; k < 128; k += 4) {
      v2f a, bm;
      // A (16x4, rows broadcast key): lanes 0-15 K={k,k+1}; lanes 16-31 K={k+2,k+3}
      a.x = keyp[k + 2 * h];
      a.y = keyp[k + 2 * h + 1];
      // B (4x16): VGPR0 = B[h, n], VGPR1 = B[h+2, n]
      bm.x = rowp[k + h];
      bm.y = rowp[k + h + 2];
      c = __builtin_amdgcn_wmma_f32_16x16x4_f32(
          /*neg_a=*/false, a, /*neg_b=*/false, bm,
          /*c_mod=*/(short)0, c, /*reuse_a=*/false, /*reuse_b=*/false);
    }
    // WAR: this buffer is the DMA target again in the next iteration; ensure our
    // DS reads have landed before new async writes can touch it.
    asm volatile("s_wait_dscnt 0x0" ::: "memory");
  }

  // C[0, n] lives in c[0] on lanes 0-15: score for seq s0+n
  if (lane < 16) {
    const int s = s0 + lane;
    float lm = __logf(mask[(size_t)s * BATCH + b]);
    wgt[(size_t)s * BATCH + b] = c[0] * 0.03125f + lm;  // 1/sqrt(1024)
  }
}

// ---------------------------------------------------------------------------
// Kernel 2: in-place softmax over s for one batch column per workgroup.
// ---------------------------------------------------------------------------
__global__ __launch_bounds__(256) void softmax_kernel(float* wgt) {
  const int b = blockIdx.x;
  const int t = threadIdx.x;
  __shared__ float red[256];

  float m = -INFINITY;
  for (int s = t; s < SEQ; s += 256) m = fmaxf(m, wgt[(size_t)s * BATCH + b]);
  red[t] = m;
  __syncthreads();
  for (int off = 128; off > 0; off >>= 1) {
    if (t < off) red[t] = fmaxf(red[t], red[t + off]);
    __syncthreads();
  }
  const float gm = red[0];
  __syncthreads();

  float sum = 0.f;
  for (int s = t; s < SEQ; s += 256) sum += __expf(wgt[(size_t)s * BATCH + b] - gm);
  red[t] = sum;
  __syncthreads();
  for (int off = 128; off > 0; off >>= 1) {
    if (t < off) red[t] += red[t + off];
    __syncthreads();
  }
  const float inv = 1.0f / red[0];

  // each s touched by exactly one thread -> race-free in-place rewrite
  for (int s = t; s < SEQ; s += 256)
    wgt[(size_t)s * BATCH + b] = __expf(wgt[(size_t)s * BATCH + b] - gm) * inv;
}

// ---------------------------------------------------------------------------
// Kernel 3: ctx[b, d0..d0+31] = sum_s dist[s,b] * enc[s,b,d]; also copies the
// `value` (cs) half of the concat output. 32 contiguous d per WG row-group:
// each global load instruction covers exactly one 128B line. 1024 WGs.
// ---------------------------------------------------------------------------
__global__ __launch_bounds__(256) void ctx_kernel(
    const float* __restrict__ enc, const float* __restrict__ cs,
    const float* __restrict__ dist, float* __restrict__ out) {
  const int b  = blockIdx.y;
  const int d0 = blockIdx.x * 32;
  const int t  = threadIdx.x;
  const int dd = t & 31;     // d lane within the 32-wide tile
  const int sg = t >> 5;     // seq sub-group 0..7

  float acc = 0.f;
  for (int s = sg; s < SEQ; s += 8)
    acc += dist[(size_t)s * BATCH + b] *
           enc[(size_t)s * (BATCH * DIM) + (size_t)b * DIM + d0 + dd];

  __shared__ float red[256];
  red[t] = acc;
  __syncthreads();
  if (sg == 0) {
    float sum = acc;
    #pragma unroll
    for (int g = 1; g < 8; ++g) sum += red[g * 32 + dd];
    out[(size_t)b * (2 * DIM) + DIM + d0 + dd] = sum;                      // ctx half
    out[(size_t)b * (2 * DIM) + d0 + dd] = cs[(size_t)b * DIM + d0 + dd]; // value half
  }
}

// ---------------------------------------------------------------------------
extern "C" void kernel_launch(void* const* d_in, const int* in_sizes, int n_in,
                              void* d_out, int out_size, void* d_ws, size_t ws_size,
                              hipStream_t stream) {
  const float* cs   = (const float*)d_in[0];  // [32, 1024]
  const float* enc  = (const float*)d_in[1];  // [2048, 32, 1024]
  const float* mask = (const float*)d_in[2];  // [2048, 32]
  // d_in[3], d_in[4] (decoder state/mask) unused: self_attention=False path.

  float* out = (float*)d_out;          // [32*2048] out ++ [2048*32] dist
  float* wgt = out + BATCH * 2 * DIM;  // dist region doubles as the weights buffer

  scores_kernel<<<dim3(SEQ / 16, BATCH), 32, 0, stream>>>(enc, cs, mask, wgt);
  softmax_kernel<<<BATCH, 256, 0, stream>>>(wgt);            // wgt -> dist in place
  ctx_kernel<<<dim3(DIM / 32, BATCH), 256, 0, stream>>>(enc, cs, wgt, out);
}